// ScatterMoEGatedMLP_33998961115695
// MI455X (gfx1250) — compile-verified
//
#include <hip/hip_runtime.h>
#include <math.h>

typedef __attribute__((ext_vector_type(2))) float v2f;
typedef __attribute__((ext_vector_type(8))) float v8f;

// Problem constants (match reference setup_inputs)
constexpr int N_TOK  = 4096;            // B*L = 2*2048
constexpr int DMODEL = 1024;
constexpr int FFN    = 2048;
constexpr int NEXP   = 8;
constexpr int TOPK   = 2;
constexpr int SLOT_CAP = N_TOK*TOPK + NEXP*64;  // 8704 (64-aligned per-expert padding)
constexpr int MTILES   = SLOT_CAP/64;           // 136

constexpr int ASTR  = 36;  // LDS stride (floats) for 64x32 A tile, row-major by token
constexpr int BTSTR = 36;  // LDS stride (floats) for 64x32 W tile, row-major by COLUMN (transposed)

// ---- CDNA5 async global->LDS copy (ASYNCcnt path, no VGPR staging) ---------
__device__ __forceinline__ void async_load_b128(uint32_t lds_off, const float* gptr) {
  asm volatile("global_load_async_to_lds_b128 %0, %1, off"
               :: "v"(lds_off), "v"((uint64_t)(uintptr_t)gptr)
               : "memory");
}
__device__ __forceinline__ void wait_async0() {
  asm volatile("s_wait_asynccnt 0x0" ::: "memory");
}
__device__ __forceinline__ uint32_t lds_off_of(const void* p) {
  return (uint32_t)(uintptr_t)p;   // generic LDS pointer: low 32 bits = LDS byte offset
}

// ---------------------------------------------------------------------------
// 1) Router: logits -> top-2 experts + normalized gates, per-expert counts
// ---------------------------------------------------------------------------
__global__ __launch_bounds__(256) void moe_router(const float* __restrict__ x,
                                                  const float* __restrict__ rw,
                                                  int* __restrict__ counts,
                                                  int* __restrict__ topk_id,
                                                  float* __restrict__ topk_w) {
  __shared__ float s_rw[NEXP * DMODEL];            // 32 KB
  for (int i = threadIdx.x; i < NEXP * DMODEL / 4; i += 256)
    ((float4*)s_rw)[i] = ((const float4*)rw)[i];
  __syncthreads();

  const int wave = threadIdx.x >> 5, lane = threadIdx.x & 31;
  const int n = blockIdx.x * 8 + wave;             // one wave per token
  const float* xr = x + (size_t)n * DMODEL;

  float acc[NEXP] = {};
  for (int d = lane; d < DMODEL; d += 32) {
    float xv = xr[d];
#pragma unroll
    for (int e = 0; e < NEXP; ++e) acc[e] += xv * s_rw[e * DMODEL + d];
  }
#pragma unroll
  for (int e = 0; e < NEXP; ++e)
#pragma unroll
    for (int off = 16; off > 0; off >>= 1)
      acc[e] += __shfl_xor(acc[e], off, 32);

  if (lane == 0) {
    int e0 = 0; float l0 = acc[0];
#pragma unroll
    for (int e = 1; e < NEXP; ++e) if (acc[e] > l0) { l0 = acc[e]; e0 = e; }
    int e1 = (e0 == 0) ? 1 : 0; float l1 = acc[e1];
#pragma unroll
    for (int e = 0; e < NEXP; ++e)
      if (e != e0 && acc[e] > l1) { l1 = acc[e]; e1 = e; }
    float t  = __expf(l1 - l0);                 // softmax denom cancels in top-2 renorm
    float w0 = 1.0f / (1.0f + t);
    float w1 = t / (1.0f + t);
    topk_id[2*n]   = e0;  topk_id[2*n+1] = e1;
    topk_w[2*n]    = w0;  topk_w[2*n+1]  = w1;
    atomicAdd(&counts[e0], 1);
    atomicAdd(&counts[e1], 1);
  }
}

// ---------------------------------------------------------------------------
// 2) 64-aligned exclusive scan of counts; tile->expert map; init padded slots
// ---------------------------------------------------------------------------
__global__ __launch_bounds__(256) void moe_scan(const int* __restrict__ counts,
                                                int* __restrict__ cursor,
                                                int* __restrict__ aoff,
                                                int* __restrict__ tile_expert,
                                                int* __restrict__ token_idx,
                                                float* __restrict__ gates) {
  for (int i = threadIdx.x; i < SLOT_CAP; i += 256) { token_idx[i] = -1; gates[i] = 0.0f; }
  if (threadIdx.x == 0) {
    int off = 0, tile = 0;
    for (int e = 0; e < NEXP; ++e) {
      aoff[e] = off;
      cursor[e] = 0;
      int t = (counts[e] + 63) >> 6;
      for (int i = 0; i < t; ++i) tile_expert[tile++] = e;
      off += t << 6;
    }
    for (; tile < MTILES; ++tile) tile_expert[tile] = -1;
  }
}

// ---------------------------------------------------------------------------
// 3) Scatter tokens into compacted per-expert slot regions
// ---------------------------------------------------------------------------
__global__ __launch_bounds__(256) void moe_scatter(const int* __restrict__ topk_id,
                                                   const float* __restrict__ topk_w,
                                                   const int* __restrict__ aoff,
                                                   int* __restrict__ cursor,
                                                   int* __restrict__ token_idx,
                                                   float* __restrict__ gates,
                                                   int* __restrict__ slot_of) {
  int n = blockIdx.x * 256 + threadIdx.x;
  if (n >= N_TOK) return;
#pragma unroll
  for (int k = 0; k < TOPK; ++k) {
    int e = topk_id[2*n + k];
    int pos = atomicAdd(&cursor[e], 1);
    int slot = aoff[e] + pos;
    token_idx[slot] = n;
    gates[slot] = topk_w[2*n + k];
    slot_of[2*n + k] = slot;
  }
}

// ---------------------------------------------------------------------------
// 4) Grouped GEMM1 + fused SiLU-gate:  H = silu(X@W1g) * (X@W1h)
//    A tile: async->LDS row-major; W tiles: VGPR-staged, stored TRANSPOSED so
//    both WMMA operand fragments are contiguous b64 LDS loads (no v_mov fixup)
// ---------------------------------------------------------------------------
__global__ __launch_bounds__(256) void moe_gemm1(const float* __restrict__ x,
                                                 const float* __restrict__ w1,
                                                 const int* __restrict__ tile_expert,
                                                 const int* __restrict__ token_idx,
                                                 float* __restrict__ Hout) {
  const int mt = blockIdx.y;
  const int e = tile_expert[mt];
  if (e < 0) return;

  __shared__ __align__(16) float As [2][64 * ASTR];   // [token][k]
  __shared__ __align__(16) float Bgt[2][64 * BTSTR];  // [col][k]  (transposed)
  __shared__ __align__(16) float Bht[2][64 * BTSTR];  // [col][k]  (transposed)
  __shared__ int s_tok[64];

  const int tid = threadIdx.x;
  if (tid < 64) s_tok[tid] = token_idx[mt * 64 + tid];
  __syncthreads();

  const int wave = tid >> 5, lane = tid & 31;
  const int mi = wave & 3, nib = (wave >> 2) << 1;
  const int col0 = blockIdx.x * 64;                   // g-column base, in [0, FFN)
  const float* w1e = w1 + (size_t)e * DMODEL * (2 * FFN);

  const uint32_t as_b[2] = { lds_off_of(&As[0][0]), lds_off_of(&As[1][0]) };

  // Pre-zero A rows of padded slots (async engine never writes them: EXEC-masked)
#pragma unroll
  for (int p = 0; p < 2; ++p) {
    int idx = p * 256 + tid;
    int row = idx >> 3, c4 = (idx & 7) << 2;
    if (s_tok[row] < 0) {
      float4 z = make_float4(0.f, 0.f, 0.f, 0.f);
      *(float4*)&As[0][row * ASTR + c4] = z;
      *(float4*)&As[1][row * ASTR + c4] = z;
    }
  }

  // async-issue A K-chunk into LDS buffer `buf`
  auto issueA = [&](int buf, int k0) {
#pragma unroll
    for (int p = 0; p < 2; ++p) {
      int idx = p * 256 + tid;
      int row = idx >> 3, c4 = (idx & 7) << 2;
      int tok = s_tok[row];
      if (tok >= 0)
        async_load_b128(as_b[buf] + (uint32_t)(row * ASTR + c4) * 4,
                        x + (size_t)tok * DMODEL + k0 + c4);
    }
  };

  // VGPR-staged W chunk: rows (tid>>4) and (tid>>4)+16, cols (tid&15)*4 .. +3
  const int wr  = tid >> 4;               // 0..15
  const int wc4 = (tid & 15) << 2;        // 0..60
  float4 bg0r, bg1r, bh0r, bh1r;
  auto loadB = [&](int k0) {
    const float* s0 = w1e + (size_t)(k0 + wr) * (2 * FFN) + col0 + wc4;
    const float* s1 = s0 + (size_t)16 * (2 * FFN);
    bg0r = *(const float4*)s0;        bh0r = *(const float4*)(s0 + FFN);
    bg1r = *(const float4*)s1;        bh1r = *(const float4*)(s1 + FFN);
  };
  auto storeB = [&](int buf) {        // transposed scatter: Bt[col][k]
    float* bg = &Bgt[buf][0];
    float* bh = &Bht[buf][0];
    bg[(wc4+0)*BTSTR + wr]      = bg0r.x;  bg[(wc4+1)*BTSTR + wr]      = bg0r.y;
    bg[(wc4+2)*BTSTR + wr]      = bg0r.z;  bg[(wc4+3)*BTSTR + wr]      = bg0r.w;
    bg[(wc4+0)*BTSTR + wr + 16] = bg1r.x;  bg[(wc4+1)*BTSTR + wr + 16] = bg1r.y;
    bg[(wc4+2)*BTSTR + wr + 16] = bg1r.z;  bg[(wc4+3)*BTSTR + wr + 16] = bg1r.w;
    bh[(wc4+0)*BTSTR + wr]      = bh0r.x;  bh[(wc4+1)*BTSTR + wr]      = bh0r.y;
    bh[(wc4+2)*BTSTR + wr]      = bh0r.z;  bh[(wc4+3)*BTSTR + wr]      = bh0r.w;
    bh[(wc4+0)*BTSTR + wr + 16] = bh1r.x;  bh[(wc4+1)*BTSTR + wr + 16] = bh1r.y;
    bh[(wc4+2)*BTSTR + wr + 16] = bh1r.z;  bh[(wc4+3)*BTSTR + wr + 16] = bh1r.w;
  };

  v8f cg0 = {}, cg1 = {}, ch0 = {}, ch1 = {};
  const int arow = mi * 16 + (lane & 15);
  const int kk   = (lane >> 4) << 1;                  // 0 | 2 per WMMA f32 layout
  const int bc0  = nib * 16 + (lane & 15);            // output column within block

  issueA(0, 0);
  loadB(0);
  wait_async0();
  storeB(0);
  __syncthreads();

  for (int k0 = 0; k0 < DMODEL; k0 += 32) {
    const int  cur  = (k0 >> 5) & 1;
    const bool more = (k0 + 32 < DMODEL);
    if (more) { issueA(cur ^ 1, k0 + 32); loadB(k0 + 32); }   // overlap with WMMA

    const float* Asc = &As [cur][0];
    const float* Bgc = &Bgt[cur][0];
    const float* Bhc = &Bht[cur][0];
#pragma unroll
    for (int ks = 0; ks < 32; ks += 4) {
      v2f a    = *(const v2f*)&Asc[arow * ASTR + ks + kk];
      v2f bg0v = *(const v2f*)&Bgc[ bc0       * BTSTR + ks + kk];
      v2f bg1v = *(const v2f*)&Bgc[(bc0 + 16) * BTSTR + ks + kk];
      v2f bh0v = *(const v2f*)&Bhc[ bc0       * BTSTR + ks + kk];
      v2f bh1v = *(const v2f*)&Bhc[(bc0 + 16) * BTSTR + ks + kk];
      cg0 = __builtin_amdgcn_wmma_f32_16x16x4_f32(false, a, false, bg0v, (short)0, cg0, false, false);
      cg1 = __builtin_amdgcn_wmma_f32_16x16x4_f32(false, a, false, bg1v, (short)0, cg1, false, false);
      ch0 = __builtin_amdgcn_wmma_f32_16x16x4_f32(false, a, false, bh0v, (short)0, ch0, false, false);
      ch1 = __builtin_amdgcn_wmma_f32_16x16x4_f32(false, a, false, bh1v, (short)0, ch1, false, false);
    }
    if (more) storeB(cur ^ 1);
    wait_async0();
    __syncthreads();
  }

  // fused activation + store H tile (C layout: vgpr i -> rows i / i+8)
  const int rbase = mt * 64 + mi * 16 + ((lane >> 4) << 3);
  const int cbase = col0 + nib * 16 + (lane & 15);
#pragma unroll
  for (int i = 0; i < 8; ++i) {
    float g0 = cg0[i], h0 = ch0[i];
    float g1 = cg1[i], h1 = ch1[i];
    float a0 = (g0 / (1.0f + __expf(-g0))) * h0;   // silu(g)*h
    float a1 = (g1 / (1.0f + __expf(-g1))) * h1;
    size_t r = (size_t)(rbase + i) * FFN;
    Hout[r + cbase]      = a0;
    Hout[r + cbase + 16] = a1;
  }
}

// ---------------------------------------------------------------------------
// 5) Grouped GEMM2 + gate:  partial[slot] = gate[slot] * (H @ W2)
// ---------------------------------------------------------------------------
__global__ __launch_bounds__(256) void moe_gemm2(const float* __restrict__ H,
                                                 const float* __restrict__ w2,
                                                 const int* __restrict__ tile_expert,
                                                 const float* __restrict__ gates,
                                                 float* __restrict__ partial) {
  const int mt = blockIdx.y;
  const int e = tile_expert[mt];
  if (e < 0) return;

  __shared__ __align__(16) float As [2][64 * ASTR];
  __shared__ __align__(16) float Bst[2][64 * BTSTR];  // [col][k] (transposed)
  __shared__ float s_gate[64];

  const int tid = threadIdx.x;
  if (tid < 64) s_gate[tid] = gates[mt * 64 + tid];

  const int wave = tid >> 5, lane = tid & 31;
  const int mi = wave & 3, nib = (wave >> 2) << 1;
  const int col0 = blockIdx.x * 64;
  const float* w2e = w2 + (size_t)e * FFN * DMODEL;
  const float* Hm  = H + (size_t)mt * 64 * FFN;

  const uint32_t as_b[2] = { lds_off_of(&As[0][0]), lds_off_of(&As[1][0]) };

  auto issueA = [&](int buf, int k0) {
#pragma unroll
    for (int p = 0; p < 2; ++p) {
      int idx = p * 256 + tid;
      int row = idx >> 3, c4 = (idx & 7) << 2;
      async_load_b128(as_b[buf] + (uint32_t)(row * ASTR + c4) * 4,
                      Hm + (size_t)row * FFN + k0 + c4);
    }
  };

  const int wr  = tid >> 4;
  const int wc4 = (tid & 15) << 2;
  float4 b0r, b1r;
  auto loadB = [&](int k0) {
    const float* s0 = w2e + (size_t)(k0 + wr) * DMODEL + col0 + wc4;
    b0r = *(const float4*)s0;
    b1r = *(const float4*)(s0 + (size_t)16 * DMODEL);
  };
  auto storeB = [&](int buf) {
    float* bs = &Bst[buf][0];
    bs[(wc4+0)*BTSTR + wr]      = b0r.x;  bs[(wc4+1)*BTSTR + wr]      = b0r.y;
    bs[(wc4+2)*BTSTR + wr]      = b0r.z;  bs[(wc4+3)*BTSTR + wr]      = b0r.w;
    bs[(wc4+0)*BTSTR + wr + 16] = b1r.x;  bs[(wc4+1)*BTSTR + wr + 16] = b1r.y;
    bs[(wc4+2)*BTSTR + wr + 16] = b1r.z;  bs[(wc4+3)*BTSTR + wr + 16] = b1r.w;
  };

  v8f c0 = {}, c1 = {};
  const int arow = mi * 16 + (lane & 15);
  const int kk   = (lane >> 4) << 1;
  const int bc0  = nib * 16 + (lane & 15);

  issueA(0, 0);
  loadB(0);
  wait_async0();
  storeB(0);
  __syncthreads();

  for (int k0 = 0; k0 < FFN; k0 += 32) {
    const int  cur  = (k0 >> 5) & 1;
    const bool more = (k0 + 32 < FFN);
    if (more) { issueA(cur ^ 1, k0 + 32); loadB(k0 + 32); }

    const float* Asc = &As [cur][0];
    const float* Bsc = &Bst[cur][0];
#pragma unroll
    for (int ks = 0; ks < 32; ks += 4) {
      v2f a   = *(const v2f*)&Asc[arow * ASTR + ks + kk];
      v2f b0v = *(const v2f*)&Bsc[ bc0       * BTSTR + ks + kk];
      v2f b1v = *(const v2f*)&Bsc[(bc0 + 16) * BTSTR + ks + kk];
      c0 = __builtin_amdgcn_wmma_f32_16x16x4_f32(false, a, false, b0v, (short)0, c0, false, false);
      c1 = __builtin_amdgcn_wmma_f32_16x16x4_f32(false, a, false, b1v, (short)0, c1, false, false);
    }
    if (more) storeB(cur ^ 1);
    wait_async0();
    __syncthreads();
  }

  const int rloc  = mi * 16 + ((lane >> 4) << 3);
  const int cbase = col0 + nib * 16 + (lane & 15);
#pragma unroll
  for (int i = 0; i < 8; ++i) {
    float gte = s_gate[rloc + i];
    size_t r = (size_t)(mt * 64 + rloc + i) * DMODEL;
    partial[r + cbase]      = gte * c0[i];
    partial[r + cbase + 16] = gte * c1[i];
  }
}

// ---------------------------------------------------------------------------
// 6) Combine: out[n] = partial[slot_of[n,0]] + partial[slot_of[n,1]]
// ---------------------------------------------------------------------------
__global__ __launch_bounds__(256) void moe_combine(const float* __restrict__ partial,
                                                   const int* __restrict__ slot_of,
                                                   float* __restrict__ out) {
  int idx = blockIdx.x * 256 + threadIdx.x;       // float4 index
  int n  = idx >> 8;                              // 1024/4 = 256 float4 per row
  int d4 = idx & 255;
  int s0 = slot_of[2*n], s1 = slot_of[2*n + 1];
  float4 a = ((const float4*)partial)[(size_t)s0 * 256 + d4];
  float4 b = ((const float4*)partial)[(size_t)s1 * 256 + d4];
  ((float4*)out)[idx] = make_float4(a.x + b.x, a.y + b.y, a.z + b.z, a.w + b.w);
}

// ---------------------------------------------------------------------------
extern "C" void kernel_launch(void* const* d_in, const int* in_sizes, int n_in,
                              void* d_out, int out_size, void* d_ws, size_t ws_size,
                              hipStream_t stream) {
  (void)in_sizes; (void)n_in; (void)out_size; (void)ws_size;
  const float* x  = (const float*)d_in[0];   // [N, D]
  const float* rw = (const float*)d_in[1];   // [E, D]
  const float* w1 = (const float*)d_in[2];   // [E, D, 2F]
  const float* w2 = (const float*)d_in[3];   // [E, F, D]
  float* out = (float*)d_out;                // [N, D]

  char* ws = (char*)d_ws;
  size_t o = 0;
  auto alloc = [&](size_t bytes) -> void* {
    o = (o + 255) & ~(size_t)255;
    void* p = ws + o;
    o += bytes;
    return p;
  };
  int*   counts    = (int*)alloc(NEXP * 4);
  int*   cursor    = (int*)alloc(NEXP * 4);
  int*   aoff      = (int*)alloc(NEXP * 4);
  int*   texp      = (int*)alloc(MTILES * 4);
  int*   topk_id   = (int*)alloc((size_t)N_TOK * 2 * 4);
  float* topk_w    = (float*)alloc((size_t)N_TOK * 2 * 4);
  int*   token_idx = (int*)alloc((size_t)SLOT_CAP * 4);
  float* gates     = (float*)alloc((size_t)SLOT_CAP * 4);
  int*   slot_of   = (int*)alloc((size_t)N_TOK * 2 * 4);
  float* Hbuf      = (float*)alloc((size_t)SLOT_CAP * FFN * 4);     // ~71 MB
  float* partial   = (float*)alloc((size_t)SLOT_CAP * DMODEL * 4);  // ~36 MB

  hipMemsetAsync(counts, 0, NEXP * 4, stream);

  moe_router <<<N_TOK / 8,   256, 0, stream>>>(x, rw, counts, topk_id, topk_w);
  moe_scan   <<<1,           256, 0, stream>>>(counts, cursor, aoff, texp, token_idx, gates);
  moe_scatter<<<N_TOK / 256, 256, 0, stream>>>(topk_id, topk_w, aoff, cursor,
                                               token_idx, gates, slot_of);

  dim3 g1(FFN / 64, MTILES);
  moe_gemm1<<<g1, 256, 0, stream>>>(x, w1, texp, token_idx, Hbuf);

  dim3 g2(DMODEL / 64, MTILES);
  moe_gemm2<<<g2, 256, 0, stream>>>(Hbuf, w2, texp, gates, partial);

  moe_combine<<<(N_TOK * DMODEL / 4) / 256, 256, 0, stream>>>(partial, slot_of, out);
}